// GraphAttentionLayer_v2_38371237823022
// MI455X (gfx1250) — compile-verified
//
#include <hip/hip_runtime.h>
#include <hip/hip_bf16.h>

#define DIM    128
#define ALPHA  0.5f
#define EPSF   1e-12f
#define THRESH 1e-11f

typedef __attribute__((ext_vector_type(2))) float v2f;
typedef __attribute__((ext_vector_type(8))) float v8f;

// ---------------------------------------------------------------------------
// Phase 1: degree accumulation.  deg_s[cs] += out_w ; deg_r[cs] += in_w
// with cs = [s, r] doubled edges, out_w = [cos^2, sin^2], in_w = [sin^2, cos^2].
// ---------------------------------------------------------------------------
__global__ void fdgcn_degree_kernel(const long long* __restrict__ snd,
                                    const long long* __restrict__ rcv,
                                    const float* __restrict__ theta,
                                    float* __restrict__ deg_s,
                                    float* __restrict__ deg_r,
                                    int E) {
  int e = blockIdx.x * blockDim.x + threadIdx.x;
  if (e >= E) return;
  int s = (int)snd[e];
  int r = (int)rcv[e];
  float sn, cn;
  __sincosf(theta[e], &sn, &cn);
  float c2 = cn * cn;
  float s2 = sn * sn;
  unsafeAtomicAdd(&deg_s[s], c2);   // forward half: cs=s, out_w=cos^2
  unsafeAtomicAdd(&deg_s[r], s2);   // reverse half: cs=r, out_w=sin^2
  unsafeAtomicAdd(&deg_r[s], s2);   // forward half: in_w=sin^2
  unsafeAtomicAdd(&deg_r[r], c2);   // reverse half: in_w=cos^2
}

// ---------------------------------------------------------------------------
// Phase 2: dinv = (deg + EPS < THRESH) ? 0 : rsqrt(deg + EPS), in place.
// ---------------------------------------------------------------------------
__global__ void fdgcn_rsqrt_kernel(float* __restrict__ deg, int n) {
  int i = blockIdx.x * blockDim.x + threadIdx.x;
  if (i >= n) return;
  float d = deg[i] + EPSF;
  deg[i] = (d < THRESH) ? 0.0f : rsqrtf(d);
}

// ---------------------------------------------------------------------------
// Phase 3: edge message scatter.  One wave32 per half-edge; each lane moves a
// float4 slice of the 128-float feature row (coalesced 512B gather, dense
// global_atomic_add_f32 scatter into two accumulators).
// ---------------------------------------------------------------------------
__global__ void fdgcn_scatter_kernel(const float* __restrict__ x,
                                     const long long* __restrict__ snd,
                                     const long long* __restrict__ rcv,
                                     const float* __restrict__ theta,
                                     const float* __restrict__ dinv_s,
                                     const float* __restrict__ dinv_r,
                                     float* __restrict__ xs2d,
                                     float* __restrict__ xd2s,
                                     int E) {
  int lane = threadIdx.x & 31;
  int idx  = blockIdx.x * (blockDim.x >> 5) + (threadIdx.x >> 5);
  if (idx >= 2 * E) return;

  bool fwd = (idx < E);
  int e = fwd ? idx : idx - E;
  int s = (int)snd[e];
  int r = (int)rcv[e];
  float sn, cn;
  __sincosf(theta[e], &sn, &cn);
  float c2 = cn * cn;
  float s2 = sn * sn;

  int   cs = fwd ? s : r;            // conv sender
  int   cr = fwd ? r : s;            // conv receiver
  float ow = fwd ? c2 : s2;          // out_w for this half-edge
  float iw = fwd ? s2 : c2;          // in_w  for this half-edge

  float w_s2t = dinv_s[cs] * ow * dinv_r[cr];
  float w_t2s = dinv_r[cs] * iw * dinv_s[cr];

  const float4 m = ((const float4*)(x + (long)cs * DIM))[lane];
  float* o1 = xs2d + (long)cr * DIM + lane * 4;
  float* o2 = xd2s + (long)cr * DIM + lane * 4;
  unsafeAtomicAdd(o1 + 0, m.x * w_s2t);
  unsafeAtomicAdd(o1 + 1, m.y * w_s2t);
  unsafeAtomicAdd(o1 + 2, m.z * w_s2t);
  unsafeAtomicAdd(o1 + 3, m.w * w_s2t);
  unsafeAtomicAdd(o2 + 0, m.x * w_t2s);
  unsafeAtomicAdd(o2 + 1, m.y * w_t2s);
  unsafeAtomicAdd(o2 + 2, m.z * w_t2s);
  unsafeAtomicAdd(o2 + 3, m.w * w_t2s);
}

// ---------------------------------------------------------------------------
// Phase 4: fused dual GEMM with native fp32 WMMA (V_WMMA_F32_16X16X4_F32).
// One wave per 16x16 output tile; 8 waves/block cover the 128 output columns
// of one 16-row stripe.  K loop = 32 steps of 4, two WMMAs per step.
//
// ISA VGPR layouts (CDNA5 ISA):
//   A 16x4 f32 : lane L holds row M = L&15; VGPR j holds K = j + 2*(L>=16)
//   B 4x16 f32 : lane L holds col N = L&15; VGPR j holds K = j + 2*(L>=16)
//   C/D 16x16  : lane L holds col N = L&15; VGPR v holds M = v + 8*(L>=16)
// ---------------------------------------------------------------------------
__global__ void fdgcn_gemm_wmma_kernel(const float* __restrict__ xs2d,
                                       const float* __restrict__ xd2s,
                                       const float* __restrict__ Ws2d,
                                       const float* __restrict__ Wd2s,
                                       const float* __restrict__ bs2d,
                                       const float* __restrict__ bd2s,
                                       float* __restrict__ out,
                                       int N) {
  const int lane  = threadIdx.x & 31;
  const int wave  = threadIdx.x >> 5;      // 0..7 -> output column tile
  const int m0    = blockIdx.x * 16;
  const int n0    = wave * 16;
  const int l15   = lane & 15;
  const int khalf = (lane >> 4) << 1;      // 0 for lanes 0-15, 2 for 16-31

  // Clamp A row for partial tiles (stores are guarded below); keeps EXEC all-1s.
  int arow = m0 + l15;
  if (arow > N - 1) arow = N - 1;
  const float* a1p = xs2d + (long)arow * DIM;
  const float* a2p = xd2s + (long)arow * DIM;

  v8f acc1 = {};
  v8f acc2 = {};

  #pragma unroll 4
  for (int k = 0; k < DIM; k += 4) {
    const int ka = k + khalf;
    v2f a1 = { a1p[ka], a1p[ka + 1] };
    v2f a2 = { a2p[ka], a2p[ka + 1] };
    v2f b1 = { Ws2d[ka * DIM + n0 + l15], Ws2d[(ka + 1) * DIM + n0 + l15] };
    v2f b2 = { Wd2s[ka * DIM + n0 + l15], Wd2s[(ka + 1) * DIM + n0 + l15] };
    acc1 = __builtin_amdgcn_wmma_f32_16x16x4_f32(false, a1, false, b1,
                                                 (short)0, acc1, false, false);
    acc2 = __builtin_amdgcn_wmma_f32_16x16x4_f32(false, a2, false, b2,
                                                 (short)0, acc2, false, false);
  }

  const float bias = ALPHA * bs2d[n0 + l15] + (1.0f - ALPHA) * bd2s[n0 + l15];
  const int rbase = (lane < 16) ? 0 : 8;
  #pragma unroll
  for (int v = 0; v < 8; ++v) {
    int m = m0 + rbase + v;
    if (m < N) {
      out[(long)m * DIM + n0 + l15] =
          ALPHA * acc1[v] + (1.0f - ALPHA) * acc2[v] + bias;
    }
  }
}

// ---------------------------------------------------------------------------
extern "C" void kernel_launch(void* const* d_in, const int* in_sizes, int n_in,
                              void* d_out, int out_size, void* d_ws, size_t ws_size,
                              hipStream_t stream) {
  const float*     x     = (const float*)d_in[0];
  const float*     Ws2d  = (const float*)d_in[1];
  const float*     bs2d  = (const float*)d_in[2];
  const float*     Wd2s  = (const float*)d_in[3];
  const float*     bd2s  = (const float*)d_in[4];
  const float*     theta = (const float*)d_in[5];
  const long long* ei    = (const long long*)d_in[6];   // int64 edge_index [2, E]

  const int E = in_sizes[5];           // theta length
  const int N = in_sizes[0] / DIM;     // node count

  const long long* snd = ei;           // edge_index[0]
  const long long* rcv = ei + E;       // edge_index[1]

  float* ws    = (float*)d_ws;
  float* deg_s = ws;                               // N floats (-> dinv_s in place)
  float* deg_r = ws + N;                           // N floats (-> dinv_r in place)
  float* xs2d  = ws + 2 * (size_t)N;               // N*DIM floats
  float* xd2s  = xs2d + (size_t)N * DIM;           // N*DIM floats

  // Zero the accumulation scratch every call (graph-capture-safe memset node).
  size_t zero_bytes = (size_t)N * (2 + 2 * DIM) * sizeof(float);
  hipMemsetAsync(d_ws, 0, zero_bytes, stream);

  // Phase 1: degree scatter-add over E edges (both half-edges per thread).
  fdgcn_degree_kernel<<<(E + 255) / 256, 256, 0, stream>>>(
      snd, rcv, theta, deg_s, deg_r, E);

  // Phase 2: in-place rsqrt normalization over both degree vectors (2N).
  fdgcn_rsqrt_kernel<<<(2 * N + 255) / 256, 256, 0, stream>>>(deg_s, 2 * N);

  // Phase 3: message scatter, one wave32 per half-edge (8 per 256-thread block).
  int halfE = 2 * E;
  fdgcn_scatter_kernel<<<(halfE + 7) / 8, 256, 0, stream>>>(
      x, snd, rcv, theta, deg_s, deg_r, xs2d, xd2s, E);

  // Phase 4: fused dual fp32 WMMA GEMM + bias + alpha blend.
  fdgcn_gemm_wmma_kernel<<<(N + 15) / 16, 256, 0, stream>>>(
      xs2d, xd2s, Ws2d, Wd2s, bs2d, bd2s, (float*)d_out, N);
}